// Loss_38508676776436
// MI455X (gfx1250) — compile-verified
//
#include <hip/hip_runtime.h>
#include <hip/hip_bf16.h>
#include <math.h>

typedef __attribute__((ext_vector_type(2))) float v2f;
typedef __attribute__((ext_vector_type(8))) float v8f;

#define NB 4
#define NPATCH 25
#define SPP 100
#define NPTS 2500            // NPATCH * SPP
#define NTGT 8192
#define MTILES 157           // ceil(NPTS / 16)
#define WAVES 8              // waves per block in kernel 1
#define GROUPS 20            // ceil(MTILES / WAVES)
#define TCHUNK 2048          // targets staged in LDS per chunk (32KB as float4)

// ---------------------------------------------------------------------------
// Kernel 1: source-side chamfer via V_WMMA_F32_16X16X4_F32.
// 256-thread blocks (8 waves); each wave owns one 16-row source tile.
// Targets are staged chunk-wise into LDS as (-2x, -2y, -2z, |t|^2) so the
// WMMA accumulator seeded with (|p|^2 + |t|^2) directly produces the squared
// distance:  D = A x (-2T) + (|p|^2 + |t|^2).  Per-lane running (min, argmin)
// in registers, then a wave32 xor-shuffle reduction over the 16 column lanes.
// ---------------------------------------------------------------------------
__global__ __launch_bounds__(256) void chamfer_src_wmma(
    const float* __restrict__ points, const float* __restrict__ normals,
    const float* __restrict__ tpoints, const float* __restrict__ tnormals,
    float* __restrict__ ca, float* __restrict__ na)
{
  __shared__ float4 s_tgt[TCHUNK];   // 32 KB of the 320 KB WGP LDS

  const int b    = blockIdx.x / GROUPS;
  const int grp  = blockIdx.x % GROUPS;
  const int tid  = threadIdx.x;
  const int wv   = tid >> 5;
  const int lane = tid & 31;
  const int lcol = lane & 15;
  const int hi   = lane >> 4;

  const bool valid = (grp * WAVES + wv) < MTILES;
  int mt = grp * WAVES + wv;
  if (mt >= MTILES) mt = MTILES - 1;       // clamped waves duplicate tile 156
  const int row0 = mt * 16;

  // ---- A fragment (16x4 f32): lanes 0-15 hold (K0,K1), lanes 16-31 (K2,K3)
  int arow = row0 + lcol;
  if (arow >= NPTS) arow = NPTS - 1;       // duplicate rows masked on write
  const float* p = points + ((size_t)b * NPTS + arow) * 3;
  const float px = p[0], py = p[1], pz = p[2];
  const float pn2 = px * px + py * py + pz * pz;
  v2f afrag;
  afrag.x = hi ? pz : px;
  afrag.y = hi ? 0.0f : py;

  // |p|^2 for the 8 D-rows this lane owns (rows r + 8*hi, held by lane r+8*hi)
  float pn2r[8];
#pragma unroll
  for (int r = 0; r < 8; ++r) pn2r[r] = __shfl(pn2, r + 8 * hi, 32);

  float best[8];
  int   bidx[8];
#pragma unroll
  for (int r = 0; r < 8; ++r) { best[r] = 3.4e38f; bidx[r] = 0; }

  const float* tb = tpoints + (size_t)b * NTGT * 3;
  for (int c0 = 0; c0 < NTGT; c0 += TCHUNK) {
    // cooperative stage of one target chunk, pre-scaled for the WMMA form
    for (int i = tid; i < TCHUNK; i += 256) {
      const float* t = tb + (size_t)(c0 + i) * 3;
      const float x = t[0], y = t[1], z = t[2];
      s_tgt[i] = make_float4(-2.0f * x, -2.0f * y, -2.0f * z,
                             x * x + y * y + z * z);
    }
    __syncthreads();

    // prefetch next chunk into cache while this one is consumed
    if (c0 + TCHUNK < NTGT) {
      for (int i = tid; i < TCHUNK; i += 256)
        __builtin_prefetch(tb + (size_t)(c0 + TCHUNK + i) * 3, 0, 1);
    }

#pragma unroll 2
    for (int j = 0; j < TCHUNK; j += 16) {
      const float4 tv = s_tgt[j + lcol];
      v2f bfrag;
      bfrag.x = hi ? tv.z : tv.x;
      bfrag.y = hi ? 0.0f : tv.y;
      v8f c;
#pragma unroll
      for (int r = 0; r < 8; ++r) c[r] = pn2r[r] + tv.w;
      const v8f d = __builtin_amdgcn_wmma_f32_16x16x4_f32(
          /*neg_a=*/false, afrag, /*neg_b=*/false, bfrag,
          /*c_mod=*/(short)0, c, /*reuse_a=*/false, /*reuse_b=*/false);
      const int colg = c0 + j + lcol;
#pragma unroll
      for (int r = 0; r < 8; ++r) {
        if (d[r] < best[r]) { best[r] = d[r]; bidx[r] = colg; }
      }
    }
    __syncthreads();
  }

  // cross-lane min over the 16 column lanes in each half (xor stays in half)
#pragma unroll
  for (int m = 1; m < 16; m <<= 1) {
#pragma unroll
    for (int r = 0; r < 8; ++r) {
      const float ov = __shfl_xor(best[r], m, 32);
      const int   oi = __shfl_xor(bidx[r], m, 32);
      if (ov < best[r] || (ov == best[r] && oi < bidx[r])) {
        best[r] = ov; bidx[r] = oi;
      }
    }
  }

  // lanes 0..7 write rows 0..7; lanes 16..23 write rows 8..15
  const int r = lcol;
  if (valid && r < 8) {
    const int n = row0 + r + 8 * hi;
    if (n < NPTS) {
      const size_t o = (size_t)b * NPTS + n;
      ca[o] = best[r];
      const float* nn = normals + o * 3;
      const float* tn = tnormals + ((size_t)b * NTGT + bidx[r]) * 3;
      const float dx = nn[0] - tn[0], dy = nn[1] - tn[1], dz = nn[2] - tn[2];
      na[o] = dx * dx + dy * dy + dz * dz;
    }
  }
}

// ---------------------------------------------------------------------------
// Kernel 2: target-side chamfer. 256 threads/block, 32 blocks per batch;
// the 2500 source points (+ precomputed |p|^2) are staged in LDS (40KB).
// ---------------------------------------------------------------------------
__global__ __launch_bounds__(256) void chamfer_tgt(
    const float* __restrict__ points, const float* __restrict__ normals,
    const float* __restrict__ tpoints, const float* __restrict__ tnormals,
    float* __restrict__ cb, float* __restrict__ nb)
{
  __shared__ float sp[NPTS * 4];
  const int b = blockIdx.x >> 5;
  const int t = ((blockIdx.x & 31) << 8) + threadIdx.x;

  const float* pb = points + (size_t)b * NPTS * 3;
  for (int n = threadIdx.x; n < NPTS; n += 256) {
    const float x = pb[n * 3], y = pb[n * 3 + 1], z = pb[n * 3 + 2];
    sp[n * 4 + 0] = x; sp[n * 4 + 1] = y; sp[n * 4 + 2] = z;
    sp[n * 4 + 3] = x * x + y * y + z * z;
  }
  __syncthreads();

  const float* tp = tpoints + ((size_t)b * NTGT + t) * 3;
  const float tx = tp[0], ty = tp[1], tz = tp[2];
  const float tn2 = tx * tx + ty * ty + tz * tz;
  float best = 3.4e38f;
  int   bi = 0;
#pragma unroll 4
  for (int n = 0; n < NPTS; ++n) {
    const float dot = sp[n * 4] * tx + sp[n * 4 + 1] * ty + sp[n * 4 + 2] * tz;
    const float d = sp[n * 4 + 3] + tn2 - 2.0f * dot;
    if (d < best) { best = d; bi = n; }
  }
  const size_t o = (size_t)b * NTGT + t;
  cb[o] = best;
  const float* tn = tnormals + o * 3;
  const float* nn = normals + ((size_t)b * NPTS + bi) * 3;
  const float dx = tn[0] - nn[0], dy = tn[1] - nn[1], dz = tn[2] - nn[2];
  nb[o] = dx * dx + dy * dy + dz * dz;
}

// ---------------------------------------------------------------------------
// Kernel 3: planar patch loss. One thread per (b, patch): weighted 3x3 normal
// equations, ridge 1e-6, Gaussian elimination with partial pivoting, weighted
// residual. 100 patches total.
// ---------------------------------------------------------------------------
__global__ __launch_bounds__(128) void planar_kernel(
    const float* __restrict__ st, const float* __restrict__ points,
    const float* __restrict__ mtds, float* __restrict__ planar_pp)
{
  const int idx = blockIdx.x * blockDim.x + threadIdx.x;
  if (idx >= NB * NPATCH) return;
  const int b = idx / NPATCH, pch = idx % NPATCH;
  const float* stp = st + (size_t)idx * SPP * 2;
  const float* pts = points + ((size_t)b * NPTS + pch * SPP) * 3;
  const float* w   = mtds + (size_t)idx * SPP;

  float m00 = 0, m01 = 0, m02 = 0, m11 = 0, m12 = 0, m22 = 0;
  float R[3][3] = {{0, 0, 0}, {0, 0, 0}, {0, 0, 0}};
  for (int s = 0; s < SPP; ++s) {
    const float u = stp[s * 2], v = stp[s * 2 + 1], wi = w[s];
    m00 += wi * u * u; m01 += wi * u * v; m02 += wi * u;
    m11 += wi * v * v; m12 += wi * v;     m22 += wi;
    const float qx = pts[s * 3], qy = pts[s * 3 + 1], qz = pts[s * 3 + 2];
    R[0][0] += wi * u * qx; R[0][1] += wi * u * qy; R[0][2] += wi * u * qz;
    R[1][0] += wi * v * qx; R[1][1] += wi * v * qy; R[1][2] += wi * v * qz;
    R[2][0] += wi * qx;     R[2][1] += wi * qy;     R[2][2] += wi * qz;
  }
  float M[3][3] = {{m00 + 1e-6f, m01, m02},
                   {m01, m11 + 1e-6f, m12},
                   {m02, m12, m22 + 1e-6f}};
  for (int k = 0; k < 3; ++k) {
    int piv = k; float mx = fabsf(M[k][k]);
    for (int j = k + 1; j < 3; ++j)
      if (fabsf(M[j][k]) > mx) { mx = fabsf(M[j][k]); piv = j; }
    if (piv != k) {
      for (int c2 = 0; c2 < 3; ++c2) {
        float tm = M[k][c2]; M[k][c2] = M[piv][c2]; M[piv][c2] = tm;
        tm = R[k][c2]; R[k][c2] = R[piv][c2]; R[piv][c2] = tm;
      }
    }
    const float inv = 1.0f / M[k][k];
    for (int j = k + 1; j < 3; ++j) {
      const float f = M[j][k] * inv;
      for (int c2 = k; c2 < 3; ++c2) M[j][c2] -= f * M[k][c2];
      for (int c2 = 0; c2 < 3; ++c2) R[j][c2] -= f * R[k][c2];
    }
  }
  float A[3][3];
  for (int k = 2; k >= 0; --k)
    for (int c2 = 0; c2 < 3; ++c2) {
      float s2 = R[k][c2];
      for (int j = k + 1; j < 3; ++j) s2 -= M[k][j] * A[j][c2];
      A[k][c2] = s2 / M[k][k];
    }
  float num = 0, den = 0;
  for (int s = 0; s < SPP; ++s) {
    const float u = stp[s * 2], v = stp[s * 2 + 1], wi = w[s];
    float err = 0;
    for (int d2 = 0; d2 < 3; ++d2) {
      const float pr = u * A[0][d2] + v * A[1][d2] + A[2][d2] - pts[s * 3 + d2];
      err += pr * pr;
    }
    num += wi * err; den += wi;
  }
  planar_pp[idx] = num / den;
}

// ---------------------------------------------------------------------------
// Kernel 4: finalize. Single block: per-b weighted chamfer/normal sums,
// per-patch template-normal sums (LDS atomics), plain sums for cb/nb, then
// the 6 output scalars.
// ---------------------------------------------------------------------------
__global__ __launch_bounds__(256) void finalize_kernel(
    const float* __restrict__ normals, const float* __restrict__ tmplN,
    const float* __restrict__ mtds, const float* __restrict__ ca,
    const float* __restrict__ na, const float* __restrict__ cb,
    const float* __restrict__ nb, const float* __restrict__ planar_pp,
    float* __restrict__ out)
{
  __shared__ float s_mtd[NB], s_ca[NB], s_na[NB];
  __shared__ float s_tnlW[NB * NPATCH], s_tnlM[NB * NPATCH];
  __shared__ float s_cb, s_nb;
  const int tid = threadIdx.x;
  if (tid < NB) { s_mtd[tid] = 0.f; s_ca[tid] = 0.f; s_na[tid] = 0.f; }
  for (int i = tid; i < NB * NPATCH; i += 256) { s_tnlW[i] = 0.f; s_tnlM[i] = 0.f; }
  if (tid == 0) { s_cb = 0.f; s_nb = 0.f; }
  __syncthreads();

  float lm[NB] = {0, 0, 0, 0}, lca[NB] = {0, 0, 0, 0}, lna[NB] = {0, 0, 0, 0};
  for (int i = tid; i < NB * NPTS; i += 256) {
    const int b = i / NPTS, n = i % NPTS;
    const float w = mtds[i];
    lm[b] += w; lca[b] += w * ca[i]; lna[b] += w * na[i];
    const float* nn = normals + (size_t)i * 3;
    const float* tn = tmplN + (size_t)n * 3;
    const float dx = tn[0] - nn[0], dy = tn[1] - nn[1], dz = tn[2] - nn[2];
    const float tnl = dx * dx + dy * dy + dz * dz;
    const int patch = b * NPATCH + n / SPP;
    atomicAdd(&s_tnlW[patch], w * tnl);
    atomicAdd(&s_tnlM[patch], w);
  }
  for (int b = 0; b < NB; ++b) {
    atomicAdd(&s_mtd[b], lm[b]);
    atomicAdd(&s_ca[b], lca[b]);
    atomicAdd(&s_na[b], lna[b]);
  }
  float lcb = 0.f, lnb = 0.f;
  for (int i = tid; i < NB * NTGT; i += 256) { lcb += cb[i]; lnb += nb[i]; }
  atomicAdd(&s_cb, lcb);
  atomicAdd(&s_nb, lnb);
  __syncthreads();

  if (tid == 0) {
    float ca_m = 0.f, na_m = 0.f;
    for (int b = 0; b < NB; ++b) {
      ca_m += s_ca[b] / s_mtd[b];
      na_m += s_na[b] / s_mtd[b];
    }
    ca_m /= (float)NB; na_m /= (float)NB;
    const float cb_m = s_cb / (float)(NB * NTGT);
    const float nb_m = s_nb / (float)(NB * NTGT);
    const float chamfer = 0.5f * (ca_m + cb_m);
    const float normals_l = 0.5f * (na_m + nb_m);
    float tmpl = 0.f, plan = 0.f;
    for (int i = 0; i < NB * NPATCH; ++i) {
      tmpl += s_tnlW[i] / s_tnlM[i];
      plan += planar_pp[i];
    }
    tmpl /= (float)(NB * NPATCH);
    plan /= (float)(NB * NPATCH);
    const float loss = chamfer + 1.0f * normals_l + 2.0f * plan + 0.1f * tmpl;
    out[0] = loss; out[1] = chamfer; out[2] = normals_l;
    out[3] = 0.0f; out[4] = plan;    out[5] = tmpl;
  }
}

// ---------------------------------------------------------------------------
extern "C" void kernel_launch(void* const* d_in, const int* in_sizes, int n_in,
                              void* d_out, int out_size, void* d_ws, size_t ws_size,
                              hipStream_t stream) {
  const float* points   = (const float*)d_in[0];  // [4, 2500, 3]
  const float* normals  = (const float*)d_in[1];  // [4, 2500, 3]
  const float* mtds     = (const float*)d_in[2];  // [4, 25, 100]
  const float* st       = (const float*)d_in[3];  // [4, 25, 100, 2]
  const float* tpoints  = (const float*)d_in[4];  // [4, 8192, 3]
  const float* tnormals = (const float*)d_in[5];  // [4, 8192, 3]
  const float* tmplN    = (const float*)d_in[6];  // [2500, 3]
  (void)in_sizes; (void)n_in; (void)ws_size; (void)out_size;

  float* ws = (float*)d_ws;
  float* ca        = ws;                       // 10000
  float* na        = ca + NB * NPTS;           // 10000
  float* cb        = na + NB * NPTS;           // 32768
  float* nb        = cb + NB * NTGT;           // 32768
  float* planar_pp = nb + NB * NTGT;           // 100

  chamfer_src_wmma<<<NB * GROUPS, 256, 0, stream>>>(points, normals, tpoints,
                                                    tnormals, ca, na);
  chamfer_tgt<<<NB * (NTGT / 256), 256, 0, stream>>>(points, normals, tpoints,
                                                     tnormals, cb, nb);
  planar_kernel<<<1, 128, 0, stream>>>(st, points, mtds, planar_pp);
  finalize_kernel<<<1, 256, 0, stream>>>(normals, tmplN, mtds, ca, na, cb, nb,
                                         planar_pp, (float*)d_out);
}